// AdditiveAttention_83322365542767
// MI455X (gfx1250) — compile-verified
//
#include <hip/hip_runtime.h>
#include <hip/hip_bf16.h>

// Exact insight: reference applies softmax over a size-1 axis -> weights are
// identically 1.0, so out[b,q,v] = sum_k values[b,k,v], broadcast over q.
// queries/keys/ws are mathematically irrelevant to the output.
//
// We compute the per-batch column sum of values with chained
// V_WMMA_F32_16X16X4_F32 ops (A = all-ones 16x4 f32, B = 4x16 f32 tile of
// values), which is exact f32 arithmetic, then broadcast-store to all Q rows.

typedef __attribute__((ext_vector_type(2))) float v2f;
typedef __attribute__((ext_vector_type(8))) float v8f;

#define BATCH 4
#define QDIM 256
#define KDIM 256
#define VDIM 256

// grid = BATCH * (VDIM/128) = 8 blocks, 256 threads (8 waves) per block.
// wave w of block (b, half) computes column sums for columns
// [half*128 + w*16, half*128 + w*16 + 16) of batch b.
__global__ void AdditiveAttention_83322365542767_kernel(
    const float* __restrict__ values,  // [B, K, V]
    float* __restrict__ out)           // [B, Q, V]
{
    const int blk   = blockIdx.x;       // 0..7
    const int b     = blk >> 1;         // batch index
    const int halfV = blk & 1;          // which 128-column half of V
    const int tid   = threadIdx.x;
    const int wave  = tid >> 5;         // 0..7
    const int lane  = tid & 31;
    const int n     = lane & 15;        // column within the 16-wide tile
    const int hi    = lane >> 4;        // lane half selects K rows

    const int colBase = halfV * 128 + wave * 16;
    const float* __restrict__ vb = values + (size_t)b * KDIM * VDIM;

    // A = all-ones 16x4 f32 matrix (layout-invariant: every element is 1.0f).
    v2f a;
    a.x = 1.0f;
    a.y = 1.0f;

    v8f acc = {};  // f32 accumulator, starts at zero

    // K reduction: 256 / 4 = 64 chained WMMAs. B tile rows k0..k0+3 are
    // distributed across the two lane halves / two B VGPRs; with an all-ones
    // A the column sum is exact regardless of the K striping, and columns map
    // as N = lane%16 per the documented B/C/D layouts.
#pragma unroll 8
    for (int k0 = 0; k0 < KDIM; k0 += 4) {
        const int r0 = k0 + hi * 2;     // rows r0 and r0+1 for this lane half
        v2f bm;
        bm.x = vb[(size_t)r0 * VDIM + colBase + n];
        bm.y = vb[(size_t)(r0 + 1) * VDIM + colBase + n];
        acc = __builtin_amdgcn_wmma_f32_16x16x4_f32(
            /*neg_a=*/false, a, /*neg_b=*/false, bm,
            /*c_mod=*/(short)0, acc, /*reuse_a=*/false, /*reuse_b=*/false);
    }

    // D VGPR0, lanes 0..15 hold D[M=0][N=0..15] = column sums for this tile.
    __shared__ float sums[128];
    if (lane < 16) {
        sums[wave * 16 + n] = acc[0];
    }
    __syncthreads();

    // Broadcast the 128 column sums of this block to all 256 query rows.
    // Thread t covers column c = t%128; two q rows per iteration, fully
    // coalesced 512B stores per half-block.
    const int c    = tid & 127;
    const int qoff = tid >> 7;          // 0 or 1
    const float s  = sums[c];
    float* __restrict__ ob = out + (size_t)b * QDIM * VDIM + halfV * 128;

#pragma unroll 4
    for (int q = qoff; q < QDIM; q += 2) {
        ob[(size_t)q * VDIM + c] = s;
    }
}

extern "C" void kernel_launch(void* const* d_in, const int* in_sizes, int n_in,
                              void* d_out, int out_size, void* d_ws, size_t ws_size,
                              hipStream_t stream) {
    // setup_inputs() order: queries[0], keys[1], values[2], ws[3].
    // Only `values` affects the output (softmax over a size-1 axis == 1.0).
    const float* values = (const float*)d_in[2];
    float* out = (float*)d_out;

    (void)in_sizes; (void)n_in; (void)d_ws; (void)ws_size; (void)out_size;

    const int blocks = BATCH * (VDIM / 128);  // 8
    AdditiveAttention_83322365542767_kernel<<<blocks, 256, 0, stream>>>(values, out);
}